// StructuredSparseTransform_25838523253063
// MI455X (gfx1250) — compile-verified
//
#include <hip/hip_runtime.h>

typedef __attribute__((ext_vector_type(2))) float        v2f;
typedef __attribute__((ext_vector_type(8))) float        v8f;
typedef __attribute__((ext_vector_type(4))) unsigned int v4u;
typedef __attribute__((ext_vector_type(8))) int          v8i;
typedef __attribute__((ext_vector_type(4))) int          v4i;

#define HIDDEN   4096
#define BLK      819           // diagonal block size (2*0.1*4096)
#define M_TOTAL  (8 * 2048)    // flattened batch*seq
#define TILE_M   64
#define TILE_N   64
#define KC       32            // K chunk per LDS stage
#define LDS_STR  36            // padded row stride (floats)
#define LDS_BUF_BYTES (TILE_M * LDS_STR * 4)

#if defined(__has_builtin)
#  if __has_builtin(__builtin_amdgcn_tensor_load_to_lds)
#    define HAVE_TDM 1
#  else
#    define HAVE_TDM 0
#  endif
#else
#  define HAVE_TDM 0
#endif

#if HAVE_TDM
// Issue one TDM DMA: 2-D tile of KC x TILE_M f32 from global (row stride
// HIDDEN elems) into LDS at lds_addr, padding +4 DWORDs after every 32
// DWORDs so LDS rows land at stride LDS_STR = 36 floats.
static __device__ __forceinline__ void tdm_load_x(unsigned lds_addr,
                                                  unsigned long long gaddr) {
    v4u g0;
    g0[0] = 1u;                                        // count=1 (valid), user mode
    g0[1] = lds_addr;                                  // LDS byte address
    g0[2] = (unsigned)(gaddr & 0xFFFFFFFFu);           // global_addr[31:0]
    g0[3] = (unsigned)((gaddr >> 32) & 0x1FFFFFFu)     // global_addr[56:32]
          | (2u << 30);                                // type = 2 ("image")
    v8i g1;
    g1[0] = (2 << 16)      // data_size = 4 bytes
          | (1 << 20)      // pad_enable
          | (4 << 22)      // pad_interval: after 32 DWORDs
          | (3 << 25);     // pad_amount: 4 DWORDs
    g1[1] = (KC & 0xFFFF) << 16;                       // tensor_dim0[15:0]
    g1[2] = (int)(((unsigned)TILE_M & 0xFFFFu) << 16); // dim0 hi=0 | tensor_dim1 lo
    g1[3] = KC << 16;                                  // dim1 hi=0 | tile_dim0 = KC
    g1[4] = TILE_M;                                    // tile_dim1 = 64, tile_dim2 = 0
    g1[5] = HIDDEN;                                    // tensor_dim0_stride lo32
    g1[6] = 0;                                         // stride hi / dim1_stride
    g1[7] = 0;
    v4i z4 = {0, 0, 0, 0};
#if __clang_major__ >= 23
    v8i z8 = {0, 0, 0, 0, 0, 0, 0, 0};
    __builtin_amdgcn_tensor_load_to_lds(g0, g1, z4, z4, z8, 0);
#else
    __builtin_amdgcn_tensor_load_to_lds(g0, g1, z4, z4, 0);
#endif
}
#endif

__global__ __launch_bounds__(256)
void bsgemm_wmma_f32(const float* __restrict__ X,
                     const float* __restrict__ W,
                     float* __restrict__ Out)
{
    __shared__ float Xs[2][TILE_M * LDS_STR];
    __shared__ float Ws[2][TILE_N * LDS_STR];

    const int tid  = threadIdx.x;
    const int lane = tid & 31;
    const int wave = tid >> 5;      // 0..7
    const int half = lane >> 4;     // 0 or 1
    const int ln   = lane & 15;     // 0..15
    const int wm   = wave >> 1;     // 0..3 -> 16-row strip
    const int wn   = wave & 1;      // 0..1 -> 32-col strip

    // grid.x = N tiles (fastest): concurrent blocks share the same X slab
    const int n_base = blockIdx.x * TILE_N;
    const int m_base = blockIdx.y * TILE_M;

    // K range = union of diagonal blocks covered by this tile's 64 output
    // columns, rounded out to KC (4096 % KC == 0 -> never OOB).
    const int jb_lo = n_base / BLK;
    const int jb_hi = (n_base + TILE_N - 1) / BLK;
    const int k_lo  = (jb_lo * BLK) & ~(KC - 1);
    int k_hi = (jb_hi + 1) * BLK;
    if (k_hi > HIDDEN) k_hi = HIDDEN;
    k_hi = (k_hi + KC - 1) & ~(KC - 1);
    const int nchunks = (k_hi - k_lo) / KC;

    v8f c0 = {};
    v8f c1 = {};

    // Loader mapping for W: 256 threads cover 32 rows x 8 float4 cols.
    const int lr = tid >> 3;        // 0..31
    const int lc = (tid & 7) << 2;  // 0,4,...,28

    const float* wp0 = W + (size_t)(n_base + lr)      * HIDDEN + lc;
    const float* wp1 = W + (size_t)(n_base + lr + 32) * HIDDEN + lc;
    const int nb0base = ((n_base + lr)      / BLK) * BLK;
    const int nb1base = ((n_base + lr + 32) / BLK) * BLK;

#if !HAVE_TDM
    const float* xp0 = X + (size_t)(m_base + lr)      * HIDDEN + lc;
    const float* xp1 = X + (size_t)(m_base + lr + 32) * HIDDEN + lc;
#endif

#if HAVE_TDM
    const unsigned long long xg_base =
        (unsigned long long)(uintptr_t)(X + (size_t)m_base * HIDDEN + k_lo);
    unsigned lds_x0 = (unsigned)(uintptr_t)&Xs[0][0];
#endif

    // ---- Prologue: stage chunk 0 into buffer 0 ----
#if HAVE_TDM
    if (wave == 0) tdm_load_x(lds_x0, xg_base);
#else
    {
        const float4 a = *(const float4*)(xp0 + k_lo);
        const float4 b = *(const float4*)(xp1 + k_lo);
        *(float4*)&Xs[0][lr * LDS_STR + lc]        = a;
        *(float4*)&Xs[0][(lr + 32) * LDS_STR + lc] = b;
    }
#endif
    {
        float4 a = *(const float4*)(wp0 + k_lo);
        float4 b = *(const float4*)(wp1 + k_lo);
        const int k = k_lo + lc;
        a.x = ((unsigned)(k     - nb0base) < BLK) ? a.x : 0.0f;
        a.y = ((unsigned)(k + 1 - nb0base) < BLK) ? a.y : 0.0f;
        a.z = ((unsigned)(k + 2 - nb0base) < BLK) ? a.z : 0.0f;
        a.w = ((unsigned)(k + 3 - nb0base) < BLK) ? a.w : 0.0f;
        b.x = ((unsigned)(k     - nb1base) < BLK) ? b.x : 0.0f;
        b.y = ((unsigned)(k + 1 - nb1base) < BLK) ? b.y : 0.0f;
        b.z = ((unsigned)(k + 2 - nb1base) < BLK) ? b.z : 0.0f;
        b.w = ((unsigned)(k + 3 - nb1base) < BLK) ? b.w : 0.0f;
        *(float4*)&Ws[0][lr * LDS_STR + lc]        = a;
        *(float4*)&Ws[0][(lr + 32) * LDS_STR + lc] = b;
    }

    // ---- Pipelined main loop: one barrier per chunk, double-buffered ----
    for (int c = 0; c < nchunks; ++c) {
        const int buf  = c & 1;
        const int nbuf = buf ^ 1;
        const int more = (c + 1 < nchunks);
        const int k1   = k_lo + (c + 1) * KC;

#if HAVE_TDM
        if (wave == 0) __builtin_amdgcn_s_wait_tensorcnt(0);
#endif
        __syncthreads();   // publish buffer `buf` (DS stores + TDM) to all waves

        // Kick off chunk c+1 into the other buffer while we compute chunk c.
        float4 wa, wb;
        if (more) {
#if HAVE_TDM
            if (wave == 0)
                tdm_load_x((unsigned)(uintptr_t)&Xs[nbuf][0],
                           xg_base + (unsigned long long)(c + 1) * (KC * 4));
#else
            const float4 xa4 = *(const float4*)(xp0 + k1);
            const float4 xb4 = *(const float4*)(xp1 + k1);
            *(float4*)&Xs[nbuf][lr * LDS_STR + lc]        = xa4;
            *(float4*)&Xs[nbuf][(lr + 32) * LDS_STR + lc] = xb4;
#endif
            wa = *(const float4*)(wp0 + k1);
            wb = *(const float4*)(wp1 + k1);
        }

        // --- Compute: V_WMMA_F32_16X16X4_F32 over chunk c ---
        const float* xa  = &Xs[buf][(wm * 16      + ln) * LDS_STR + 2 * half];
        const float* wb0 = &Ws[buf][(wn * 32      + ln) * LDS_STR + 2 * half];
        const float* wb1 = &Ws[buf][(wn * 32 + 16 + ln) * LDS_STR + 2 * half];
        #pragma unroll
        for (int kk = 0; kk < KC; kk += 4) {
            const v2f a  = *(const v2f*)(xa  + kk);
            const v2f b0 = *(const v2f*)(wb0 + kk);
            const v2f b1 = *(const v2f*)(wb1 + kk);
            c0 = __builtin_amdgcn_wmma_f32_16x16x4_f32(
                     false, a, false, b0, (short)0, c0, false, false);
            c1 = __builtin_amdgcn_wmma_f32_16x16x4_f32(
                     false, a, false, b1, (short)0, c1, false, false);
        }

        // Stage chunk c+1's W tile (masked) into the other buffer.
        if (more) {
            const int k = k1 + lc;
            wa.x = ((unsigned)(k     - nb0base) < BLK) ? wa.x : 0.0f;
            wa.y = ((unsigned)(k + 1 - nb0base) < BLK) ? wa.y : 0.0f;
            wa.z = ((unsigned)(k + 2 - nb0base) < BLK) ? wa.z : 0.0f;
            wa.w = ((unsigned)(k + 3 - nb0base) < BLK) ? wa.w : 0.0f;
            wb.x = ((unsigned)(k     - nb1base) < BLK) ? wb.x : 0.0f;
            wb.y = ((unsigned)(k + 1 - nb1base) < BLK) ? wb.y : 0.0f;
            wb.z = ((unsigned)(k + 2 - nb1base) < BLK) ? wb.z : 0.0f;
            wb.w = ((unsigned)(k + 3 - nb1base) < BLK) ? wb.w : 0.0f;
            *(float4*)&Ws[nbuf][lr * LDS_STR + lc]        = wa;
            *(float4*)&Ws[nbuf][(lr + 32) * LDS_STR + lc] = wb;
        }
    }

    // --- Epilogue: C/D layout -> lane ln = col, VGPR r = row r + 8*half ---
    const int out_n0 = n_base + wn * 32 + ln;
    #pragma unroll
    for (int r = 0; r < 8; ++r) {
        const int m = m_base + wm * 16 + half * 8 + r;
        float* orow = Out + (size_t)m * HIDDEN;
        orow[out_n0]      = c0[r];
        orow[out_n0 + 16] = c1[r];
    }
}

extern "C" void kernel_launch(void* const* d_in, const int* in_sizes, int n_in,
                              void* d_out, int out_size, void* d_ws, size_t ws_size,
                              hipStream_t stream)
{
    const float* X = (const float*)d_in[0];   // [8,2048,4096] flattened
    const float* W = (const float*)d_in[1];   // [4096,4096]
    // d_in[2] (mask) recomputed analytically on device; not needed.
    float* Out = (float*)d_out;

    dim3 grid(HIDDEN / TILE_N, M_TOTAL / TILE_M);   // x = N tiles (fastest)
    hipLaunchKernelGGL(bsgemm_wmma_f32, grid, dim3(256), 0, stream, X, W, Out);
}